// MultiLabelSupConLoss_21620865368546
// MI455X (gfx1250) — compile-verified
//
#include <hip/hip_runtime.h>
#include <hip/hip_bf16.h>
#include <math.h>

// Problem constants (from reference setup_inputs)
#define BATCH   4096
#define NVIEW   2
#define DIM     128
#define LDIM    100
#define VB      (BATCH * NVIEW)   // 8192
#define INV_T   (1.0f / 0.07f)

typedef __attribute__((ext_vector_type(2))) float v2f;
typedef __attribute__((ext_vector_type(8))) float v8f;

// Native fp32 WMMA: D(16x16,f32) = A(16x4,f32) * B(4x16,f32) + C
static __device__ __forceinline__ v8f wmma_f32(v2f a, v2f b, v8f c) {
    return __builtin_amdgcn_wmma_f32_16x16x4_f32(
        /*neg_a=*/false, a, /*neg_b=*/false, b,
        /*c_mod=*/(short)0, c, /*reuse_a=*/false, /*reuse_b=*/false);
}

#define ZERO8 {0.f, 0.f, 0.f, 0.f, 0.f, 0.f, 0.f, 0.f}

// ---------------------------------------------------------------------------
// Kernel 0: per-row label sums  rs[i] = sum_k labels[i][k]
// ---------------------------------------------------------------------------
__global__ void labels_rowsum_kernel(const float* __restrict__ lab,
                                     float* __restrict__ rs) {
    int r = blockIdx.x * blockDim.x + threadIdx.x;
    if (r < BATCH) {
        const float* p = lab + (size_t)r * LDIM;
        float s = 0.0f;
        #pragma unroll 4
        for (int k = 0; k < LDIM; ++k) s += p[k];
        rs[r] = s;
    }
}

// ---------------------------------------------------------------------------
// Kernel 1: rowmax[i] = max_{j<VB} dot(features[i,0,:], contrast[j,:])
// One block (8 waves) per 16-row i-tile; each wave does 2 adjacent j-tiles
// per iteration (A registers reused, 4 independent WMMA chains).
// A-layout (16x4 f32): lane l -> row M = l&15, holds K = 2*(l>>4) + {0,1}
// B-layout (4x16 f32): lane l -> col N = l&15, holds K = 2*(l>>4) + {0,1}
// C/D layout: lane l -> col N = l&15, VGPR v -> row M = v + 8*(l>>4)
// ---------------------------------------------------------------------------
__global__ void rowmax_kernel(const float* __restrict__ F,
                              float* __restrict__ rowmax) {
    const int i0   = blockIdx.x * 16;
    const int lane = threadIdx.x & 31;
    const int wave = threadIdx.x >> 5;      // 0..7
    const int m    = lane & 15;             // row (A) / col (B,C)
    const int h    = lane >> 4;             // half select
    const int koff = 2 * h;

    // Hoist the full A operand (this lane's half-row) into registers.
    const float* arow = F + (size_t)(i0 + m) * (NVIEW * DIM);  // view-0 row
    v2f av[DIM / 4];
    #pragma unroll
    for (int kk = 0; kk < DIM / 4; ++kk)
        av[kk] = *(const v2f*)(arow + kk * 4 + koff);

    float rm[8];
    #pragma unroll
    for (int v = 0; v < 8; ++v) rm[v] = -INFINITY;

    // 512 j-tiles; each wave takes pairs: 32 iterations of 2 tiles.
    for (int jt = 2 * wave; jt < VB / 16; jt += 16) {
        const int jj0 = jt * 16 + m;
        const int jj1 = jj0 + 16;
        const float* brow0 =
            F + (size_t)(((jj0 & (BATCH - 1)) << 1) + (jj0 >> 12)) * DIM;
        const float* brow1 =
            F + (size_t)(((jj1 & (BATCH - 1)) << 1) + (jj1 >> 12)) * DIM;

        // Stage both B tiles (distinct regs -> batched loads, few waits).
        v2f b0[DIM / 4], b1[DIM / 4];
        #pragma unroll
        for (int kk = 0; kk < DIM / 4; ++kk) {
            b0[kk] = *(const v2f*)(brow0 + kk * 4 + koff);
            b1[kk] = *(const v2f*)(brow1 + kk * 4 + koff);
        }

        // 4 independent accumulation chains (2 tiles x 2 k-partials).
        v8f a0 = ZERO8, a1 = ZERO8, a2 = ZERO8, a3 = ZERO8;
        #pragma unroll
        for (int kk = 0; kk < DIM / 4; kk += 2) {
            a0 = wmma_f32(av[kk],     b0[kk],     a0);
            a1 = wmma_f32(av[kk + 1], b0[kk + 1], a1);
            a2 = wmma_f32(av[kk],     b1[kk],     a2);
            a3 = wmma_f32(av[kk + 1], b1[kk + 1], a3);
        }
        #pragma unroll
        for (int v = 0; v < 8; ++v) {
            rm[v] = fmaxf(rm[v], a0[v] + a1[v]);
            rm[v] = fmaxf(rm[v], a2[v] + a3[v]);
        }
    }

    // max across the 16 lanes of each half (columns of the tile)
    #pragma unroll
    for (int off = 1; off < 16; off <<= 1) {
        #pragma unroll
        for (int v = 0; v < 8; ++v)
            rm[v] = fmaxf(rm[v], __shfl_xor(rm[v], off, 32));
    }

    __shared__ float lmax[8][16];
    if (m == 0) {
        #pragma unroll
        for (int v = 0; v < 8; ++v) lmax[wave][v + 8 * h] = rm[v];
    }
    __syncthreads();
    if (threadIdx.x < 16) {
        float mx = lmax[0][threadIdx.x];
        #pragma unroll
        for (int w = 1; w < 8; ++w) mx = fmaxf(mx, lmax[w][threadIdx.x]);
        rowmax[i0 + threadIdx.x] = mx;
    }
}

// ---------------------------------------------------------------------------
// Kernel 2: fused feature-dot + label-intersection tiles -> per-row loss terms
// ---------------------------------------------------------------------------
__global__ void loss_rows_kernel(const float* __restrict__ F,
                                 const float* __restrict__ Lab,
                                 const float* __restrict__ rowmax,
                                 const float* __restrict__ rs,
                                 float* __restrict__ per_row,
                                 float* __restrict__ has_pos) {
    const int i0   = blockIdx.x * 16;
    const int lane = threadIdx.x & 31;
    const int wave = threadIdx.x >> 5;
    const int m    = lane & 15;
    const int h    = lane >> 4;
    const int koff = 2 * h;

    // Hoisted A operands (feature half-row and label half-row).
    const float* aFrow = F   + (size_t)(i0 + m) * (NVIEW * DIM);  // view-0 row
    const float* aLrow = Lab + (size_t)(i0 + m) * LDIM;
    v2f aF[DIM / 4];
    #pragma unroll
    for (int kk = 0; kk < DIM / 4; ++kk)
        aF[kk] = *(const v2f*)(aFrow + kk * 4 + koff);
    v2f aL[LDIM / 4];
    #pragma unroll
    for (int kk = 0; kk < LDIM / 4; ++kk)
        aL[kk] = *(const v2f*)(aLrow + kk * 4 + koff);

    float rm[8], rsi[8];
    #pragma unroll
    for (int v = 0; v < 8; ++v) {
        rm[v]  = rowmax[i0 + v + 8 * h];   // raw-dot row max (full row)
        rsi[v] = rs[i0 + v + 8 * h];
    }

    float pos[8], tot[8], cnt[8];
    #pragma unroll
    for (int v = 0; v < 8; ++v) { pos[v] = 0.f; tot[v] = 0.f; cnt[v] = 0.f; }

    for (int jt = wave; jt < BATCH / 16; jt += 8) {
        const int jj = jt * 16 + m;                    // j < BATCH => view 0
        const float  rsj = rs[jj];

        // ---- feature-dot tile (staged loads, 2 k-partial chains) ----
        const float* bFrow = F + (size_t)jj * (NVIEW * DIM);
        v8f d0 = ZERO8, d1 = ZERO8;
        {
            v2f bF[DIM / 4];
            #pragma unroll
            for (int kk = 0; kk < DIM / 4; ++kk)
                bF[kk] = *(const v2f*)(bFrow + kk * 4 + koff);
            #pragma unroll
            for (int kk = 0; kk < DIM / 4; kk += 2) {
                d0 = wmma_f32(aF[kk],     bF[kk],     d0);
                d1 = wmma_f32(aF[kk + 1], bF[kk + 1], d1);
            }
        }

        // ---- label-intersection tile (staged loads, 2 k-partial chains) ----
        const float* bLrow = Lab + (size_t)jj * LDIM;
        v8f i0v = ZERO8, i1v = ZERO8;
        {
            v2f bL[LDIM / 4];
            #pragma unroll
            for (int kk = 0; kk < LDIM / 4; ++kk)
                bL[kk] = *(const v2f*)(bLrow + kk * 4 + koff);
            #pragma unroll
            for (int kk = 0; kk + 1 < LDIM / 4; kk += 2) {
                i0v = wmma_f32(aL[kk],     bL[kk],     i0v);
                i1v = wmma_f32(aL[kk + 1], bL[kk + 1], i1v);
            }
            i0v = wmma_f32(aL[LDIM / 4 - 1], bL[LDIM / 4 - 1], i0v); // kk=24
        }

        #pragma unroll
        for (int v = 0; v < 8; ++v) {
            float dot   = d0[v] + d1[v];
            float e     = __expf((dot - rm[v]) * INV_T);
            float inter = i0v[v] + i1v[v];
            float uni   = rsi[v] + rsj - inter;
            // sim >= 0.5  <=>  inter >= 0.5*(union + 1e-6)   (denominator > 0)
            bool  p     = inter >= 0.5f * (uni + 1e-6f);
            tot[v] += e;
            pos[v] += p ? e   : 0.f;
            cnt[v] += p ? 1.f : 0.f;
        }
    }

    // sum across the 16 lanes of each half
    #pragma unroll
    for (int off = 1; off < 16; off <<= 1) {
        #pragma unroll
        for (int v = 0; v < 8; ++v) {
            pos[v] += __shfl_xor(pos[v], off, 32);
            tot[v] += __shfl_xor(tot[v], off, 32);
            cnt[v] += __shfl_xor(cnt[v], off, 32);
        }
    }

    __shared__ float lred[8][16][3];
    if (m == 0) {
        #pragma unroll
        for (int v = 0; v < 8; ++v) {
            const int r = v + 8 * h;
            lred[wave][r][0] = pos[v];
            lred[wave][r][1] = tot[v];
            lred[wave][r][2] = cnt[v];
        }
    }
    __syncthreads();
    if (threadIdx.x < 16) {
        const int r = threadIdx.x;
        float posS = 0.f, totS = 0.f, cntS = 0.f;
        #pragma unroll
        for (int w = 0; w < 8; ++w) {
            posS += lred[w][r][0];
            totS += lred[w][r][1];
            cntS += lred[w][r][2];
        }
        bool  has = cntS > 0.5f;
        // -log(pos/den) = log(den) - log(pos); den = pos_sum + neg_sum = totS
        float pr  = has ? (__logf(totS) - __logf(posS)) : 0.f;
        per_row[i0 + r] = pr;
        has_pos[i0 + r] = has ? 1.f : 0.f;
    }
}

// ---------------------------------------------------------------------------
// Kernel 3: deterministic scalar reduction -> loss
// ---------------------------------------------------------------------------
__global__ void finalize_kernel(const float* __restrict__ per_row,
                                const float* __restrict__ has_pos,
                                float* __restrict__ out) {
    __shared__ float sl[256], sc[256];
    float s = 0.f, c = 0.f;
    for (int r = threadIdx.x; r < BATCH; r += 256) {
        s += per_row[r];
        c += has_pos[r];
    }
    sl[threadIdx.x] = s;
    sc[threadIdx.x] = c;
    __syncthreads();
    for (int o = 128; o > 0; o >>= 1) {
        if (threadIdx.x < o) {
            sl[threadIdx.x] += sl[threadIdx.x + o];
            sc[threadIdx.x] += sc[threadIdx.x + o];
        }
        __syncthreads();
    }
    if (threadIdx.x == 0) out[0] = sl[0] / fmaxf(sc[0], 1.f);
}

// ---------------------------------------------------------------------------
extern "C" void kernel_launch(void* const* d_in, const int* in_sizes, int n_in,
                              void* d_out, int out_size, void* d_ws, size_t ws_size,
                              hipStream_t stream) {
    (void)in_sizes; (void)n_in; (void)out_size; (void)ws_size;

    const float* features = (const float*)d_in[0];  // [4096, 2, 128] f32
    const float* labels   = (const float*)d_in[1];  // [4096, 100]    f32
    float* out = (float*)d_out;

    float* ws      = (float*)d_ws;
    float* rowmax  = ws;                 // 4096
    float* rs      = ws + BATCH;         // 4096
    float* per_row = ws + 2 * BATCH;     // 4096
    float* has_pos = ws + 3 * BATCH;     // 4096

    labels_rowsum_kernel<<<BATCH / 256, 256, 0, stream>>>(labels, rs);
    rowmax_kernel<<<BATCH / 16, 256, 0, stream>>>(features, rowmax);
    loss_rows_kernel<<<BATCH / 16, 256, 0, stream>>>(features, labels,
                                                     rowmax, rs,
                                                     per_row, has_pos);
    finalize_kernel<<<1, 256, 0, stream>>>(per_row, has_pos, out);
}